// HGT_18975165514373
// MI455X (gfx1250) — compile-verified
//
#include <hip/hip_runtime.h>
#include <hip/hip_bf16.h>

#define NNODES 200000
#define NEDGES 600000
#define HID 128
#define NH 8
#define HD 16
#define NL 2
#define MR 64  // rows per block (4 x 16 sub-tiles per wave)

typedef __attribute__((ext_vector_type(2))) float v2f;
typedef __attribute__((ext_vector_type(8))) float v8f;

// ---------------------------------------------------------------------------
// WMMA fp32 GEMM:  C[M x NC] = post( A[M x K] @ W[K x NC] + bias )
// Block = NC/16 waves (wave32); each wave owns one 16-wide column tile and
// 4 M-sub-tiles (64 rows), so each B fragment feeds 4 WMMAs (4x less W
// traffic; W stays resident in WGP$ while A/C stream at HBM rate).
// PRE: 0 = none, 1 = exact gelu on A while staging to LDS
// POST: 0 = bias, 1 = bias+relu, 2 = bias then sigmoid-skip mix with Xold
// ---------------------------------------------------------------------------
template <int K, int NC, int PRE, int POST>
__global__ __launch_bounds__(NC * 2) void gemm_wmma_kernel(
    const float* __restrict__ A, const float* __restrict__ W,
    const float* __restrict__ bias, float* __restrict__ Cout,
    const float* __restrict__ Xold, const float* __restrict__ skipPtr) {
  __shared__ float Atile[MR][K + 1];  // +1 pad breaks LDS bank conflicts
  const int tid = threadIdx.x;
  const int lane = tid & 31;
  const int wave = tid >> 5;
  const long mbase = (long)blockIdx.x * MR;

  // cooperative A-tile load (MR x K) with optional gelu pre-activation
  for (int idx = tid; idx < MR * (K / 4); idx += NC * 2) {
    const int r = idx / (K / 4);
    const int c = (idx % (K / 4)) * 4;
    const float4 vv = *(const float4*)(A + (mbase + r) * K + c);
    float vals[4] = {vv.x, vv.y, vv.z, vv.w};
#pragma unroll
    for (int j = 0; j < 4; ++j) {
      float x = vals[j];
      if (PRE == 1) x = 0.5f * x * (1.0f + erff(x * 0.70710678118654752f));
      Atile[r][c + j] = x;
    }
  }
  __syncthreads();

  const int row16 = lane & 15;
  const int klo = (lane >> 4) * 2;  // lanes 0-15: K+0/1, lanes 16-31: K+2/3
  const int col = wave * 16 + (lane & 15);

  v8f acc[4];
#pragma unroll
  for (int m = 0; m < 4; ++m) acc[m] = (v8f){0.f, 0.f, 0.f, 0.f, 0.f, 0.f, 0.f, 0.f};

#pragma unroll
  for (int k = 0; k < K; k += 4) {
    v2f b;
    b.x = W[(long)(k + klo) * NC + col];
    b.y = W[(long)(k + klo + 1) * NC + col];
#pragma unroll
    for (int m = 0; m < 4; ++m) {
      v2f a;
      a.x = Atile[m * 16 + row16][k + klo];
      a.y = Atile[m * 16 + row16][k + klo + 1];
      acc[m] = __builtin_amdgcn_wmma_f32_16x16x4_f32(false, a, false, b,
                                                     (short)0, acc[m], false,
                                                     false);
    }
  }

  // C/D layout: VGPR i -> (M=i, N=lane) lanes 0-15 ; (M=8+i, N=lane-16)
  const int rbase = (lane < 16) ? 0 : 8;
  float sig = 1.0f, oneminus = 0.0f;
  if (POST == 2) {
    const float sv = *skipPtr;
    sig = 1.0f / (1.0f + __expf(-sv));
    oneminus = 1.0f - sig;
  }
  const float bcol = bias[col];
#pragma unroll
  for (int m = 0; m < 4; ++m) {
#pragma unroll
    for (int i = 0; i < 8; ++i) {
      const long row = mbase + m * 16 + rbase + i;
      float v = acc[m][i] + bcol;
      if (POST == 1) v = fmaxf(v, 0.0f);
      if (POST == 2) v = sig * v + oneminus * Xold[row * NC + col];
      Cout[row * NC + col] = v;
    }
  }
}

// ---------------------------------------------------------------------------
// Fold per-head relation transform into effective weights:
//   Weff[:,16h:16h+16] = Wsrc[:,16h:16h+16] @ rel[h]
// One wave per (rowtile 0..7, head, {l,t,which}); fully unpredicated.
// ---------------------------------------------------------------------------
__global__ __launch_bounds__(32) void fold_rel_kernel(
    const float* __restrict__ wk, const float* __restrict__ wv,
    const float* __restrict__ a_rel, const float* __restrict__ m_rel,
    float* __restrict__ Wkr, float* __restrict__ Wvr) {
  const int rt = blockIdx.x;    // 0..7
  const int head = blockIdx.y;  // 0..7
  const int z = blockIdx.z;     // lt*2 + which
  const int lt = z >> 1, which = z & 1;
  const float* wsrc = (which ? wv : wk) + (size_t)lt * HID * HID;
  const float* rel =
      (which ? m_rel : a_rel) + (size_t)lt * NH * HD * HD + (size_t)head * HD * HD;
  float* wdst = (which ? Wvr : Wkr) + (size_t)lt * HID * HID;

  const int lane = threadIdx.x;
  const int r16 = lane & 15;
  const int klo = (lane >> 4) * 2;
  const int colBase = head * 16;
  const int arow = rt * 16 + r16;

  v8f acc = {0.f, 0.f, 0.f, 0.f, 0.f, 0.f, 0.f, 0.f};
#pragma unroll
  for (int k = 0; k < HD; k += 4) {
    const int kk0 = k + klo, kk1 = kk0 + 1;
    v2f a, b;
    a.x = wsrc[arow * HID + colBase + kk0];
    a.y = wsrc[arow * HID + colBase + kk1];
    b.x = rel[kk0 * HD + (lane & 15)];
    b.y = rel[kk1 * HD + (lane & 15)];
    acc = __builtin_amdgcn_wmma_f32_16x16x4_f32(false, a, false, b, (short)0,
                                                acc, false, false);
  }
  const int col = colBase + (lane & 15);
  const int rbase = (lane < 16) ? 0 : 8;
#pragma unroll
  for (int i = 0; i < 8; ++i) wdst[(rt * 16 + rbase + i) * HID + col] = acc[i];
}

// Folded biases: beff[lt,which][c] = sum_d bsrc[h*16+d] * rel[h][d][c&15]
// (tiny: 8*128 outputs, pure VALU)
__global__ __launch_bounds__(128) void fold_bias_kernel(
    const float* __restrict__ bk, const float* __restrict__ bv,
    const float* __restrict__ a_rel, const float* __restrict__ m_rel,
    float* __restrict__ bkr, float* __restrict__ bvr) {
  const int z = blockIdx.x;  // lt*2 + which
  const int lt = z >> 1, which = z & 1;
  const int c = threadIdx.x;  // 0..127
  const int h = c >> 4, e = c & 15;
  const float* bsrc = (which ? bv : bk) + (size_t)lt * HID;
  const float* rel =
      (which ? m_rel : a_rel) + (size_t)lt * NH * HD * HD + (size_t)h * HD * HD;
  float s = 0.f;
#pragma unroll
  for (int d = 0; d < HD; ++d) s += bsrc[h * HD + d] * rel[d * HD + e];
  ((which ? bvr : bkr) + (size_t)lt * HID)[c] = s;
}

// ---------------------------------------------------------------------------
// Edge phase: score -> segment max (monotone-uint atomicMax) -> exp/segment
// sum -> weighted message aggregation.  8 consecutive threads = 8 heads of one
// edge so q/kr node rows (512B) are read coalesced.
// ---------------------------------------------------------------------------
__device__ __forceinline__ unsigned enc_f32(float f) {
  const unsigned b = __float_as_uint(f);
  return (b & 0x80000000u) ? ~b : (b | 0x80000000u);
}
__device__ __forceinline__ float dec_f32(unsigned e) {
  const unsigned b = (e & 0x80000000u) ? (e ^ 0x80000000u) : ~e;
  return __uint_as_float(b);
}

__global__ void score_kernel(const float* __restrict__ q,
                             const float* __restrict__ kr,
                             const int* __restrict__ edge,
                             const float* __restrict__ p_rel,
                             float* __restrict__ score,
                             unsigned* __restrict__ menc) {
  const long idx = (long)blockIdx.x * blockDim.x + threadIdx.x;
  if (idx >= (long)NEDGES * NH) return;
  const long e = idx >> 3;
  const int h = (int)(idx & 7);
  const long si = edge[e];
  const long di = edge[NEDGES + e];
  const float* qa = q + di * HID + h * HD;
  const float* ka = kr + si * HID + h * HD;
  float s = 0.f;
#pragma unroll
  for (int j = 0; j < HD; j += 4) {
    const float4 qv = *(const float4*)(qa + j);
    const float4 kv = *(const float4*)(ka + j);
    s += qv.x * kv.x + qv.y * kv.y + qv.z * kv.z + qv.w * kv.w;
  }
  s = s * p_rel[h] * 0.25f;  // 1/sqrt(16)
  score[idx] = s;
  atomicMax(&menc[di * NH + h], enc_f32(s));
}

__global__ void exp_kernel(const int* __restrict__ edge,
                           const unsigned* __restrict__ menc,
                           float* __restrict__ score,
                           float* __restrict__ sums) {
  const long idx = (long)blockIdx.x * blockDim.x + threadIdx.x;
  if (idx >= (long)NEDGES * NH) return;
  const long e = idx >> 3;
  const int h = (int)(idx & 7);
  const long di = edge[NEDGES + e];
  float m = dec_f32(menc[di * NH + h]);  // menc==0 decodes to NaN -> 0 guard
  if (!isfinite(m)) m = 0.f;
  const float ex = __expf(score[idx] - m);
  score[idx] = ex;
  atomicAdd(&sums[di * NH + h], ex);
}

__global__ void aggregate_kernel(const int* __restrict__ edge,
                                 const float* __restrict__ ex,
                                 const float* __restrict__ sums,
                                 const float* __restrict__ vr,
                                 float* __restrict__ msg) {
  const long idx = (long)blockIdx.x * blockDim.x + threadIdx.x;
  if (idx >= (long)NEDGES * HID) return;
  const long e = idx >> 7;
  const int c = (int)(idx & 127);
  const int h = c >> 4;
  const long si = edge[e];
  const long di = edge[NEDGES + e];
  const float alpha = ex[e * NH + h] / (sums[di * NH + h] + 1e-16f);
  atomicAdd(&msg[di * HID + c], vr[si * HID + c] * alpha);
}

// ---------------------------------------------------------------------------
extern "C" void kernel_launch(void* const* d_in, const int* in_sizes, int n_in,
                              void* d_out, int out_size, void* d_ws,
                              size_t ws_size, hipStream_t stream) {
  const float* x_cpd = (const float*)d_in[0];
  const float* x_ko = (const float*)d_in[1];
  const int* edge_c2k = (const int*)d_in[2];
  const int* edge_k2c = (const int*)d_in[3];
  const float* w_in_cpd = (const float*)d_in[4];
  const float* b_in_cpd = (const float*)d_in[5];
  const float* w_in_ko = (const float*)d_in[6];
  const float* b_in_ko = (const float*)d_in[7];
  const float* w_k = (const float*)d_in[8];
  const float* w_q = (const float*)d_in[9];
  const float* w_v = (const float*)d_in[10];
  const float* w_a = (const float*)d_in[11];
  const float* b_k = (const float*)d_in[12];
  const float* b_q = (const float*)d_in[13];
  const float* b_v = (const float*)d_in[14];
  const float* b_a = (const float*)d_in[15];
  const float* a_rel = (const float*)d_in[16];
  const float* m_rel = (const float*)d_in[17];
  const float* p_rel = (const float*)d_in[18];
  const float* skip = (const float*)d_in[19];
  const float* w_out = (const float*)d_in[20];
  const float* b_out = (const float*)d_in[21];
  float* out = (float*)d_out;

  // ---- workspace layout (floats) ----
  const size_t NF = (size_t)NNODES * HID;  // 25.6M
  float* ws = (float*)d_ws;
  float* xp[2] = {ws + 0 * NF, ws + 1 * NF};
  float* qb[2] = {ws + 2 * NF, ws + 3 * NF};
  float* krb[2] = {ws + 4 * NF, ws + 5 * NF};
  float* vrb[2] = {ws + 6 * NF, ws + 7 * NF};
  float* mp[2] = {ws + 8 * NF, ws + 9 * NF};
  float* score = ws + 10 * NF;                              // E*8
  float* sums = score + (size_t)NEDGES * NH;                // N*8
  unsigned* menc = (unsigned*)(sums + (size_t)NNODES * NH); // N*8
  float* Wkr = (float*)(menc + (size_t)NNODES * NH);        // 4 * 128*128
  float* Wvr = Wkr + 4 * HID * HID;
  float* bkr = Wvr + 4 * HID * HID;  // 4 * 128
  float* bvr = bkr + 4 * HID;

  const int MT = NNODES / MR;  // 3125 row blocks (exact)
  const int EH_BLOCKS = (int)(((long)NEDGES * NH + 255) / 256);
  const int EC_BLOCKS = (int)(((long)NEDGES * HID + 255) / 256);

  // 1) fold relation transforms into effective K/V weights + biases
  fold_rel_kernel<<<dim3(8, NH, 8), 32, 0, stream>>>(w_k, w_v, a_rel, m_rel,
                                                     Wkr, Wvr);
  fold_bias_kernel<<<8, 128, 0, stream>>>(b_k, b_v, a_rel, m_rel, bkr, bvr);

  // 2) input projections with relu
  gemm_wmma_kernel<64, 128, 0, 1><<<MT, 256, 0, stream>>>(
      x_cpd, w_in_cpd, b_in_cpd, xp[0], nullptr, nullptr);
  gemm_wmma_kernel<128, 128, 0, 1><<<MT, 256, 0, stream>>>(
      x_ko, w_in_ko, b_in_ko, xp[1], nullptr, nullptr);

  const int* edges[2] = {edge_c2k, edge_k2c};

  for (int l = 0; l < NL; ++l) {
    // Q / Kr / Vr projections (relation transform pre-folded into weights)
    for (int t = 0; t < 2; ++t) {
      const size_t wo = (size_t)(l * 2 + t) * HID * HID;
      const size_t bo = (size_t)(l * 2 + t) * HID;
      gemm_wmma_kernel<128, 128, 0, 0><<<MT, 256, 0, stream>>>(
          xp[t], w_q + wo, b_q + bo, qb[t], nullptr, nullptr);
      gemm_wmma_kernel<128, 128, 0, 0><<<MT, 256, 0, stream>>>(
          xp[t], Wkr + wo, bkr + bo, krb[t], nullptr, nullptr);
      gemm_wmma_kernel<128, 128, 0, 0><<<MT, 256, 0, stream>>>(
          xp[t], Wvr + wo, bvr + bo, vrb[t], nullptr, nullptr);
    }
    // edge phase per relation (src type = r, dst type = 1-r)
    for (int r = 0; r < 2; ++r) {
      const int dt = 1 - r;
      hipMemsetAsync(menc, 0, (size_t)NNODES * NH * sizeof(unsigned), stream);
      hipMemsetAsync(sums, 0, (size_t)NNODES * NH * sizeof(float), stream);
      hipMemsetAsync(mp[dt], 0, NF * sizeof(float), stream);
      score_kernel<<<EH_BLOCKS, 256, 0, stream>>>(
          qb[dt], krb[r], edges[r], p_rel + (size_t)(l * 2 + r) * NH, score,
          menc);
      exp_kernel<<<EH_BLOCKS, 256, 0, stream>>>(edges[r], menc, score, sums);
      aggregate_kernel<<<EC_BLOCKS, 256, 0, stream>>>(edges[r], score, sums,
                                                      vrb[r], mp[dt]);
    }
    // node update: x_new = sig(skip)*(gelu(msg)@w_a + b_a) + (1-sig)*x
    // written in place over msg (block stages its rows to LDS before writing)
    for (int t = 0; t < 2; ++t) {
      const size_t wo = (size_t)(l * 2 + t) * HID * HID;
      const size_t bo = (size_t)(l * 2 + t) * HID;
      gemm_wmma_kernel<128, 128, 1, 2><<<MT, 256, 0, stream>>>(
          mp[t], w_a + wo, b_a + bo, mp[t], xp[t], skip + (l * 2 + t));
    }
    // ping-pong: msg buffers become next layer's x
    for (int t = 0; t < 2; ++t) {
      float* tmp = xp[t];
      xp[t] = mp[t];
      mp[t] = tmp;
    }
  }

  // 3) output projection for both node types, concatenated in d_out
  gemm_wmma_kernel<128, 64, 0, 0><<<MT, 128, 0, stream>>>(
      xp[0], w_out, b_out, out, nullptr, nullptr);
  gemm_wmma_kernel<128, 64, 0, 0><<<MT, 128, 0, stream>>>(
      xp[1], w_out, b_out, out + (size_t)NNODES * 64, nullptr, nullptr);
}